// ReactorNet_8856222564879
// MI455X (gfx1250) — compile-verified
//
#include <hip/hip_runtime.h>

// ReactorNet implicit-Euler + Newton, CDNA5 (gfx1250, wave32).
// One workgroup (128 thr = 4 waves) per reactor. Batched finite-difference
// Jacobian -> dense f32 GEMMs on V_WMMA_F32_16X16X4_F32.
// Stoichiometry is pre-padded into global scratch so GEMM A-fragments are
// unconditional aligned b64 loads; Arrhenius tables and logC padding are
// zero-extended so the rate epilogue is branch-free (no exec-mask churn).

#define SNUM 53      // species
#define DNUM 54      // state dim [T, Y...]
#define NCOL 55      // 54 FD columns + 1 base column
#define RXN  325     // reactions
#define RXP  336     // reactions padded to 21*16
#define RKP  336     // GEMM2 K padded
#define NPAD 64      // padded N (4 x 16 tiles)
#define KP1  56      // S padded to multiple of 4 (GEMM1 K)
#define NTHREADS 128

typedef __attribute__((ext_vector_type(2))) float v2f;
typedef __attribute__((ext_vector_type(4))) float v4f;
typedef __attribute__((ext_vector_type(8))) float v8f;

__device__ __forceinline__ v8f wmma_f32_4(v2f a, v2f b, v8f c) {
  // D = A(16x4) * B(4x16) + C, fp32
  return __builtin_amdgcn_wmma_f32_16x16x4_f32(false, a, false, b,
                                               (short)0, c, false, false);
}

// One-time per-launch padding of the constant stoichiometry:
//   pNuf [RXP][KP1]  = nuf zero-padded              (GEMM1 A operand)
//   pNet [NPAD][RKP] = (nub - nuf)^T zero-padded    (GEMM2 A operand)
__global__ void prep_kernel(const float* __restrict__ nuf,
                            const float* __restrict__ nub,
                            float* __restrict__ pNuf,
                            float* __restrict__ pNet) {
  int i = blockIdx.x * 256 + threadIdx.x;
  if (i < RXP * KP1) {
    int r = i / KP1, k = i % KP1;
    pNuf[i] = (r < RXN && k < SNUM) ? nuf[r * SNUM + k] : 0.0f;
  }
  int j = i - RXP * KP1;
  if (j >= 0 && j < NPAD * RKP) {
    int s = j / RKP, r = j % RKP;
    pNet[j] = (s < SNUM && r < RXN) ? (nub[r * SNUM + s] - nuf[r * SNUM + s])
                                    : 0.0f;
  }
}

template <bool PREP>
__launch_bounds__(NTHREADS)
__global__ void reactor_kernel(const float* __restrict__ T0,
                               const float* __restrict__ P,
                               const float* __restrict__ Y0,
                               const float* __restrict__ W,
                               const float* __restrict__ nlo,
                               const float* __restrict__ nhi,
                               const float* __restrict__ Tmid,
                               const float* __restrict__ aA,
                               const float* __restrict__ ab,
                               const float* __restrict__ aEa,
                               const float* __restrict__ pA1,   // padded nuf (or raw nuf)
                               const float* __restrict__ pA2,   // padded net^T (or raw nub)
                               const float* __restrict__ nufRaw,
                               const int*   __restrict__ nstepsPtr,
                               float* __restrict__ out)
{
  // LDS pool with phase-disjoint aliasing:
  //   sB1 [56][64] logC  (aliased later by F [54][64])
  //   sR  [64][64] rate chunk (aliased later by M [54][56] augmented)
  //   sWD [64][64] wdot per (species, column)
  __shared__ __align__(16) float sPool[KP1*NPAD + NPAD*NPAD + NPAD*NPAD];
  float* sB1 = sPool;
  float* sR  = sPool + KP1*NPAD;
  float* sWD = sR + NPAD*NPAD;
  float* sF  = sB1;   // alias: F written after GEMMs consumed logC
  float* sM  = sR;    // alias: M written after GEMM2 consumed R

  __shared__ float sYn[DNUM], sYp[DNUM], sX[DNUM], sEps[DNUM];
  __shared__ float sCp[2][SNUM], sH[2][SNUM];   // thermo at T_base / T_pert
  __shared__ float sRho[NCOL], sCpm[NCOL];
  __shared__ float sTc[2], sLnT[2], sInvT[2];
  __shared__ float sW[SNUM], sIW[SNUM];         // W, 1/W
  __shared__ __align__(16) float sAa[RXP];      // Arrhenius A   (zero-padded)
  __shared__ __align__(16) float sAb[RXP];      // Arrhenius b   (zero-padded)
  __shared__ __align__(16) float sEaR[RXP];     // Ea/R          (zero-padded)

  const int tid  = threadIdx.x;
  const int lane = tid & 31;
  const int wv   = tid >> 5;     // 4 waves
  const int hh   = (lane >> 4);  // lane half
  const int lr   = lane & 15;    // row-in-half
  const int b    = blockIdx.x;

  const float RGAS = 8.314462618f;
  const float INVR = 1.0f / RGAS;
  const float Pv = P[0];
  const int   ns = *nstepsPtr;
  const float dt = 1.0e-6f / (float)ns;

  // stage per-launch constants (zero-padded so hot loops need no guards)
  for (int i = tid; i < RXP; i += NTHREADS) {
    int  ic = (i < RXN) ? i : 0;
    bool v  = (i < RXN);
    sAa[i]  = v ? aA[ic]         : 0.0f;
    sAb[i]  = v ? ab[ic]         : 0.0f;
    sEaR[i] = v ? aEa[ic] * INVR : 0.0f;
  }
  if (tid < SNUM) { float w = W[tid]; sW[tid] = w; sIW[tid] = 1.0f / w; }
  if (tid == 0) { sYn[0] = T0[b]; sYp[0] = T0[b]; }
  if (tid < SNUM) { float v = Y0[b*SNUM + tid]; sYn[1+tid] = v; sYp[1+tid] = v; }
  __syncthreads();

  for (int st = 0; st < ns; ++st) {
    for (int nw = 0; nw < 2; ++nw) {
      // ---------------- Phase A: scalars, thermo, per-column state ----------
      if (tid < DNUM) sEps[tid] = fmaxf(fabsf(sYn[tid]) * 5.0e-4f, 1.0e-7f);
      __syncthreads();
      if (tid == 0) {
        float Tb = fminf(fmaxf(sYn[0],           200.0f), 5000.0f);
        float Tp = fminf(fmaxf(sYn[0] + sEps[0], 200.0f), 5000.0f);
        sTc[0] = Tb;  sTc[1] = Tp;
        sLnT[0] = __logf(Tb);  sLnT[1] = __logf(Tp);
        sInvT[0] = 1.0f / Tb;  sInvT[1] = 1.0f / Tp;
      }
      __syncthreads();
      if (tid < 2*SNUM) {           // NASA-7 cp/R and h/RT at the 2 temps
        int tsel = tid / SNUM, s = tid % SNUM;
        float T = sTc[tsel];
        const float* a = (T < Tmid[s]) ? (nlo + 7*s) : (nhi + 7*s);
        float T2 = T*T, T3 = T2*T, T4 = T2*T2;
        sCp[tsel][s] = a[0] + a[1]*T + a[2]*T2 + a[3]*T3 + a[4]*T4;
        sH[tsel][s]  = a[0] + a[1]*T*0.5f + a[2]*T2*(1.0f/3.0f)
                     + a[3]*T3*0.25f + a[4]*T4*0.2f + a[5]/T;
      }
      // zero logC padding: K-rows 53..55 (full width) and columns 55..63
      for (int idx = tid; idx < (KP1 - SNUM)*NPAD; idx += NTHREADS)
        sB1[(SNUM + idx / NPAD)*NPAD + (idx % NPAD)] = 0.0f;
      for (int idx = tid; idx < SNUM*(NPAD - NCOL); idx += NTHREADS)
        sB1[(idx / (NPAD - NCOL))*NPAD + NCOL + (idx % (NPAD - NCOL))] = 0.0f;
      __syncthreads();
      if (tid < NCOL) {             // one thread per FD column
        int col = tid;
        int tsel = (col == 0) ? 1 : 0;     // only column 0 perturbs T
        float T = sTc[tsel];
        float invW = 0.0f;
        for (int s = 0; s < SNUM; ++s) {
          float y = sYn[1+s] + ((col == s+1) ? sEps[col] : 0.0f);
          y = fminf(fmaxf(y, 0.0f), 1.0f);
          invW += y * sIW[s];
        }
        float rho = Pv / (RGAS * T * invW);
        float cpm = 0.0f;
        for (int s = 0; s < SNUM; ++s) {
          float y = sYn[1+s] + ((col == s+1) ? sEps[col] : 0.0f);
          y = fminf(fmaxf(y, 0.0f), 1.0f);
          float Ci = rho * y * sIW[s];
          sB1[s*NPAD + col] = __logf(Ci + 1e-30f);
          cpm += y * sCp[tsel][s] * RGAS * sIW[s];
        }
        sRho[col] = rho;  sCpm[col] = cpm;
      }
      __syncthreads();

      // hoist Newton-invariant scalars to registers
      const float lnT0 = sLnT[0], lnT1 = sLnT[1];
      const float iT0  = sInvT[0], iT1 = sInvT[1];

      // ---------------- Phase B/C: chunked GEMM1 -> rates -> GEMM2 ----------
      v8f c2[4];                        // persistent wdot accumulators
      #pragma unroll
      for (int q = 0; q < 4; ++q) {
        #pragma unroll
        for (int e = 0; e < 8; ++e) c2[q][e] = 0.0f;
      }

      for (int ch = 0; ch < 6; ++ch) { // 6 chunks of 64 reaction rows (21 mtiles)
        int rxBase = ch * 64;
        int rem = 21 - ch*4;
        int mloc = (rem < 4) ? rem : 4;

        if (PREP && ch < 5)            // hint next A-panel into cache
          __builtin_prefetch(pA1 + (size_t)(rxBase + 64) * KP1 + tid * 28, 0, 0);

        // GEMM1: LR = nuf x logC(LDS); then r = A*exp(LR + b lnT - Ea/(R T))
        for (int t = wv; t < mloc*4; t += 4) {
          int mt = t >> 2, nt = t & 3;
          v8f acc;
          #pragma unroll
          for (int e = 0; e < 8; ++e) acc[e] = 0.0f;
          int arow = rxBase + mt*16 + lr;
          #pragma unroll 7
          for (int ks = 0; ks < KP1/4; ++ks) {
            int kk = ks*4;
            int kA = kk + 2*hh;
            v2f a, bb;
            if (PREP) {
              a = *reinterpret_cast<const v2f*>(pA1 + (size_t)arow*KP1 + kA);
            } else {
              int ar  = (arow < RXN) ? arow : (RXN-1);
              int k0  = (kA   < SNUM) ? kA   : (SNUM-1);
              int k1  = (kA+1 < SNUM) ? kA+1 : (SNUM-1);
              float x0 = pA1[ar*SNUM + k0], x1 = pA1[ar*SNUM + k1];
              a.x = (arow < RXN && kA   < SNUM) ? x0 : 0.0f;
              a.y = (arow < RXN && kA+1 < SNUM) ? x1 : 0.0f;
            }
            bb.x = sB1[(kk + 2*hh    )*NPAD + nt*16 + lr];
            bb.y = sB1[(kk + 2*hh + 1)*NPAD + nt*16 + lr];
            acc = wmma_f32_4(a, bb, acc);
          }
          // branch-free rate epilogue (tables zero-padded; padding cols finite)
          int cl   = nt*16 + lr;
          float lnTv  = (cl == 0) ? lnT1 : lnT0;
          float invTv = (cl == 0) ? iT1  : iT0;
          int rx0 = rxBase + mt*16 + 8*hh;          // multiple of 8 -> aligned
          v4f aa0 = *reinterpret_cast<const v4f*>(&sAa[rx0]);
          v4f aa1 = *reinterpret_cast<const v4f*>(&sAa[rx0 + 4]);
          v4f ab0 = *reinterpret_cast<const v4f*>(&sAb[rx0]);
          v4f ab1 = *reinterpret_cast<const v4f*>(&sAb[rx0 + 4]);
          v4f ee0 = *reinterpret_cast<const v4f*>(&sEaR[rx0]);
          v4f ee1 = *reinterpret_cast<const v4f*>(&sEaR[rx0 + 4]);
          #pragma unroll
          for (int e = 0; e < 8; ++e) {
            float aav = (e < 4) ? aa0[e] : aa1[e-4];
            float abv = (e < 4) ? ab0[e] : ab1[e-4];
            float eev = (e < 4) ? ee0[e] : ee1[e-4];
            float ex  = acc[e] + abv*lnTv - eev*invTv;
            sR[(mt*16 + 8*hh + e)*NPAD + cl] = aav * __expf(ex);
          }
        }
        __syncthreads();

        // GEMM2 partial: WD += nu_net^T x R(LDS)
        int ksteps = mloc * 4;                     // only valid chunk rows
        int srow = wv*16 + lr;                     // wave owns species tile wv
        #pragma unroll
        for (int nt2 = 0; nt2 < 4; ++nt2) {
          v8f acc = c2[nt2];
          #pragma unroll 4
          for (int ks = 0; ks < ksteps; ++ks) {
            int kk = ks*4;
            int kg = rxBase + kk + 2*hh;
            v2f a, bb;
            if (PREP) {
              a = *reinterpret_cast<const v2f*>(pA2 + (size_t)srow*RKP + kg);
            } else {
              bool mok = (srow < SNUM);
              int sc  = mok ? srow : 0;
              int g0  = (kg   < RXN) ? kg   : (RXN-1);
              int g1  = (kg+1 < RXN) ? kg+1 : (RXN-1);
              float x0 = pA2[g0*SNUM + sc] - nufRaw[g0*SNUM + sc];
              float x1 = pA2[g1*SNUM + sc] - nufRaw[g1*SNUM + sc];
              a.x = (mok && kg   < RXN) ? x0 : 0.0f;
              a.y = (mok && kg+1 < RXN) ? x1 : 0.0f;
            }
            bb.x = sR[(kk + 2*hh    )*NPAD + nt2*16 + lr];
            bb.y = sR[(kk + 2*hh + 1)*NPAD + nt2*16 + lr];
            acc = wmma_f32_4(a, bb, acc);
          }
          c2[nt2] = acc;
        }
        __syncthreads();                           // WAR: next chunk rewrites sR
      }

      // spill wdot tiles to LDS
      #pragma unroll
      for (int nt2 = 0; nt2 < 4; ++nt2) {
        #pragma unroll
        for (int e = 0; e < 8; ++e)
          sWD[(wv*16 + 8*hh + e)*NPAD + nt2*16 + lr] = c2[nt2][e];
      }
      __syncthreads();

      // ---------------- Phase D: assemble f per column, build M = I - dt J ---
      if (tid < NCOL) {
        int col = tid;
        int tsel = (col == 0) ? 1 : 0;
        float T = sTc[tsel];
        float iRho = 1.0f / sRho[col];
        float sh = 0.0f;
        for (int s = 0; s < SNUM; ++s) {
          float wd = sWD[s*NPAD + col];
          sh += sH[tsel][s] * RGAS * T * wd;          // h_mol * wdot
          sF[(1+s)*NPAD + col] = wd * sW[s] * iRho;   // dY
        }
        sF[0*NPAD + col] = -sh * iRho / sCpm[col];    // dT
      }
      __syncthreads();
      for (int idx = tid; idx < DNUM*(DNUM+1); idx += NTHREADS) {
        int i = idx / (DNUM+1), j = idx % (DNUM+1);
        if (j < DNUM) {
          float Jij = (sF[i*NPAD + j] - sF[i*NPAD + (NCOL-1)]) / sEps[j];
          sM[i*KP1 + j] = ((i == j) ? 1.0f : 0.0f) - dt * Jij;
        } else {  // G = yn - y - dt f(yn), as augmented column
          sM[i*KP1 + DNUM] = sYn[i] - sYp[i] - dt * sF[i*NPAD + (NCOL-1)];
        }
      }
      __syncthreads();

      // ---------------- Phase E: single-wave GE solve (no block barriers) ----
      // LDS ops within one wave complete in order (DScnt), so wave 0 can run
      // the whole 54x54 elimination + back-substitution barrier-free.
      // M ~= I (dt ~ 6e-8) => diagonally dominant, no pivoting needed.
      if (wv == 0) {
        for (int k = 0; k < DNUM-1; ++k) {
          float pivInv = 1.0f / sM[k*KP1 + k];        // broadcast read
          for (int i = k + 1 + lane; i < DNUM; i += 32) {
            float fac = sM[i*KP1 + k] * pivInv;
            for (int j = k + 1; j <= DNUM; ++j)
              sM[i*KP1 + j] -= fac * sM[k*KP1 + j];
          }
        }
        for (int k = DNUM-1; k >= 0; --k) {
          float x = sM[k*KP1 + DNUM] / sM[k*KP1 + k]; // broadcast
          if (lane == 0) sX[k] = x;
          for (int i = lane; i < k; i += 32)
            sM[i*KP1 + DNUM] -= sM[i*KP1 + k] * x;
        }
      }
      __syncthreads();
      if (tid < DNUM) sYn[tid] -= sX[tid];
      __syncthreads();
    } // Newton
    if (tid < DNUM) sYp[tid] = sYn[tid];
    __syncthreads();
  } // steps

  if (tid < DNUM) out[b*DNUM + tid] = sYn[tid];
}

extern "C" void kernel_launch(void* const* d_in, const int* in_sizes, int n_in,
                              void* d_out, int out_size, void* d_ws, size_t ws_size,
                              hipStream_t stream) {
  (void)n_in; (void)out_size;
  const int B = in_sizes[0];
  const float* T0v  = (const float*)d_in[0];
  const float* Pv   = (const float*)d_in[1];
  const float* Y0v  = (const float*)d_in[2];
  const float* Wv   = (const float*)d_in[3];
  const float* nlo  = (const float*)d_in[4];
  const float* nhi  = (const float*)d_in[5];
  const float* Tmid = (const float*)d_in[6];
  const float* aA   = (const float*)d_in[7];
  const float* ab   = (const float*)d_in[8];
  const float* aEa  = (const float*)d_in[9];
  const float* nuf  = (const float*)d_in[10];
  const float* nub  = (const float*)d_in[11];
  const int*   nsp  = (const int*)  d_in[12];

  const size_t need = (size_t)(RXP*KP1 + NPAD*RKP) * sizeof(float);
  if (d_ws != nullptr && ws_size >= need) {
    float* pNuf = (float*)d_ws;
    float* pNet = pNuf + RXP*KP1;
    const int total = RXP*KP1 + NPAD*RKP;
    prep_kernel<<<dim3((total + 255) / 256), dim3(256), 0, stream>>>(
        nuf, nub, pNuf, pNet);
    reactor_kernel<true><<<dim3(B), dim3(NTHREADS), 0, stream>>>(
        T0v, Pv, Y0v, Wv, nlo, nhi, Tmid, aA, ab, aEa,
        pNuf, pNet, nuf, nsp, (float*)d_out);
  } else {
    reactor_kernel<false><<<dim3(B), dim3(NTHREADS), 0, stream>>>(
        T0v, Pv, Y0v, Wv, nlo, nhi, Tmid, aA, ab, aEa,
        nuf, nub, nuf, nsp, (float*)d_out);
  }
}